// MultiHeadTemporalAttention_89309549953781
// MI455X (gfx1250) — compile-verified
//
#include <hip/hip_runtime.h>
#include <hip/hip_bf16.h>

#define BATCH 4
#define SEQ   1024
#define DMODEL 1024
#define NHEADS 16
#define DK    64

typedef __attribute__((ext_vector_type(2))) float v2f;
typedef __attribute__((ext_vector_type(8))) float v8f;

// ---- WMMA f32 16x16x4 fragment loaders -------------------------------------
// A-frag (16x4, row-major A): lane<16 -> K={k0,k0+1}; lane>=16 -> K={k0+2,k0+3}
__device__ __forceinline__ v2f ld_a(const float* __restrict__ A, int lda,
                                    int row, int k0, int hlf) {
  const float* p = A + (size_t)row * lda + k0 + 2 * hlf;
  v2f a; a.x = p[0]; a.y = p[1];
  return a;
}
// B-frag (4x16) from row-major B[k][n]: VGPR0 -> K = k0+2*hlf, VGPR1 -> +1
__device__ __forceinline__ v2f ld_b_nt(const float* __restrict__ B, int ldb,
                                       int k0, int col, int hlf) {
  const float* p = B + (size_t)(k0 + 2 * hlf) * ldb + col;
  v2f b; b.x = p[0]; b.y = p[ldb];
  return b;
}
// B-frag where logical B[k][n] = Bm[n*ldb + k]  (e.g. X @ W^T, Q @ K^T)
__device__ __forceinline__ v2f ld_b_t(const float* __restrict__ Bm, int ldb,
                                      int k0, int col, int hlf) {
  const float* p = Bm + (size_t)col * ldb + k0 + 2 * hlf;
  v2f b; b.x = p[0]; b.y = p[1];
  return b;
}

// ---- One wave computes a 32x64 tile of C = A * B ---------------------------
template <bool TRANSB>
__device__ __forceinline__ void wave_gemm_32x64(const float* __restrict__ A, int lda,
                                                const float* __restrict__ B, int ldb,
                                                int row0, int col0, int Kdim,
                                                v8f (&acc)[2][4]) {
  const int lane = threadIdx.x & 31;
  const int hlf  = lane >> 4;
  const int mn   = lane & 15;
#pragma unroll
  for (int i = 0; i < 2; ++i)
#pragma unroll
    for (int j = 0; j < 4; ++j) acc[i][j] = (v8f){};

  for (int k0 = 0; k0 < Kdim; k0 += 4) {
    // keep the A stream warm (emits global_prefetch_b8 on gfx1250)
    __builtin_prefetch(A + (size_t)(row0 + mn) * lda + k0 + 64, 0, 0);

    v2f a0 = ld_a(A, lda, row0 + mn,      k0, hlf);
    v2f a1 = ld_a(A, lda, row0 + 16 + mn, k0, hlf);
    v2f b[4];
#pragma unroll
    for (int j = 0; j < 4; ++j) {
      if constexpr (TRANSB)
        b[j] = ld_b_t(B, ldb, k0, col0 + j * 16 + mn, hlf);
      else
        b[j] = ld_b_nt(B, ldb, k0, col0 + j * 16 + mn, hlf);
    }
#pragma unroll
    for (int j = 0; j < 4; ++j) {
      acc[0][j] = __builtin_amdgcn_wmma_f32_16x16x4_f32(
          false, a0, false, b[j], (short)0, acc[0][j], false, false);
      acc[1][j] = __builtin_amdgcn_wmma_f32_16x16x4_f32(
          false, a1, false, b[j], (short)0, acc[1][j], false, false);
    }
  }
}

// ---- Kernel 1: fused Q/K/V projections, output (B,H,S,dk) ------------------
__global__ void __launch_bounds__(256) qkv_kernel(
    const float* __restrict__ q, const float* __restrict__ k, const float* __restrict__ v,
    const float* __restrict__ wq, const float* __restrict__ bq,
    const float* __restrict__ wk, const float* __restrict__ bk,
    const float* __restrict__ wv, const float* __restrict__ bv,
    float* __restrict__ Qb, float* __restrict__ Kb, float* __restrict__ Vb) {
  const int gid = blockIdx.x * 8 + (threadIdx.x >> 5);
  const int mat = gid / 2048;             // 0:Q 1:K 2:V
  const int t   = gid % 2048;             // 128 row-tiles x 16 col-tiles
  const int row0 = (t / 16) * 32;
  const int col0 = (t % 16) * 64;
  const float* X    = (mat == 0) ? q  : (mat == 1) ? k  : v;
  const float* W    = (mat == 0) ? wq : (mat == 1) ? wk : wv;
  const float* bias = (mat == 0) ? bq : (mat == 1) ? bk : bv;
  float* dst        = (mat == 0) ? Qb : (mat == 1) ? Kb : Vb;

  v8f acc[2][4];
  wave_gemm_32x64<true>(X, DMODEL, W, DMODEL, row0, col0, DMODEL, acc);

  const int lane = threadIdx.x & 31, hlf = lane >> 4, mn = lane & 15;
#pragma unroll
  for (int i = 0; i < 2; ++i)
#pragma unroll
    for (int j = 0; j < 4; ++j)
#pragma unroll
      for (int r = 0; r < 8; ++r) {
        const int m = row0 + i * 16 + hlf * 8 + r;     // token index b*S+s
        const int c = col0 + j * 16 + mn;              // channel h*dk+d
        const int b = m >> 10, s = m & 1023;
        const int h = c >> 6,  d = c & 63;
        dst[(((size_t)(b * NHEADS + h)) * SEQ + s) * DK + d] = acc[i][j][r] + bias[c];
      }
}

// ---- Kernel 2: scores = Q K^T * inv_scale + bias[h] ------------------------
__global__ void __launch_bounds__(256) scores_kernel(
    const float* __restrict__ Qb, const float* __restrict__ Kb,
    const float* __restrict__ tbias, const float* __restrict__ atemp,
    float* __restrict__ attn) {
  const int gid = blockIdx.x * 8 + (threadIdx.x >> 5);
  const int bh  = gid / 512;              // 64 (b,h) pairs
  const int t   = gid % 512;              // 32 row-tiles x 16 col-tiles
  const int row0 = (t / 16) * 32;
  const int col0 = (t % 16) * 64;
  const float* A = Qb + (size_t)bh * SEQ * DK;
  const float* B = Kb + (size_t)bh * SEQ * DK;   // B[k][n] = K[n][k]

  v8f acc[2][4];
  wave_gemm_32x64<true>(A, DK, B, DK, row0, col0, DK, acc);

  const float inv = 1.0f / (8.0f * atemp[0]);    // sqrt(dk)=8, TEMPERATURE=1
  const float tb  = tbias[bh & 15];
  float* out = attn + (size_t)bh * SEQ * SEQ;

  const int lane = threadIdx.x & 31, hlf = lane >> 4, mn = lane & 15;
#pragma unroll
  for (int i = 0; i < 2; ++i)
#pragma unroll
    for (int j = 0; j < 4; ++j)
#pragma unroll
      for (int r = 0; r < 8; ++r) {
        const int m = row0 + i * 16 + hlf * 8 + r;
        const int n = col0 + j * 16 + mn;
        out[(size_t)m * SEQ + n] = acc[i][j][r] * inv + tb;
      }
}

// ---- Kernel 3: row softmax in place (rows of 1024) -------------------------
__global__ void __launch_bounds__(256) softmax_kernel(float* __restrict__ attn) {
  __shared__ float red[256];
  float* p = attn + (size_t)blockIdx.x * SEQ;
  const int t = threadIdx.x;
  float x[4], mx = -3.4e38f;
#pragma unroll
  for (int i = 0; i < 4; ++i) { x[i] = p[t + i * 256]; mx = fmaxf(mx, x[i]); }
  red[t] = mx; __syncthreads();
  for (int s = 128; s > 0; s >>= 1) { if (t < s) red[t] = fmaxf(red[t], red[t + s]); __syncthreads(); }
  mx = red[0]; __syncthreads();
  float sum = 0.f;
#pragma unroll
  for (int i = 0; i < 4; ++i) { x[i] = expf(x[i] - mx); sum += x[i]; }
  red[t] = sum; __syncthreads();
  for (int s = 128; s > 0; s >>= 1) { if (t < s) red[t] += red[t + s]; __syncthreads(); }
  const float inv = 1.0f / red[0];
#pragma unroll
  for (int i = 0; i < 4; ++i) p[t + i * 256] = x[i] * inv;
}

// ---- Kernel 4: ctx = attn @ V, stored with the torch-bug permutation -------
__global__ void __launch_bounds__(256) ctx_kernel(
    const float* __restrict__ attn, const float* __restrict__ Vb,
    float* __restrict__ Cx) {
  const int gid = blockIdx.x * 8 + (threadIdx.x >> 5);
  const int bh  = gid / 32;
  const int row0 = (gid % 32) * 32;      // N = 64 = one 64-wide tile
  const float* A = attn + (size_t)bh * SEQ * SEQ;
  const float* B = Vb   + (size_t)bh * SEQ * DK;

  v8f acc[2][4];
  wave_gemm_32x64<false>(A, SEQ, B, DK, row0, 0, SEQ, acc);

  const int b = bh >> 4, h = bh & 15;
  const int lane = threadIdx.x & 31, hlf = lane >> 4, mn = lane & 15;
#pragma unroll
  for (int i = 0; i < 2; ++i)
#pragma unroll
    for (int j = 0; j < 4; ++j)
#pragma unroll
      for (int r = 0; r < 8; ++r) {
        const int s = row0 + i * 16 + hlf * 8 + r;
        const int d = j * 16 + mn;
        // flat((1,H,B,S,dk)) viewed as (B,S,D): identical flat index
        Cx[(((size_t)(h * BATCH + b)) * SEQ + s) * DK + d] = acc[i][j][r];
      }
}

// ---- Kernel 5: out = ctx @ wo^T + bo + query (residual) --------------------
__global__ void __launch_bounds__(256) oproj_kernel(
    const float* __restrict__ Cx, const float* __restrict__ wo,
    const float* __restrict__ bo, const float* __restrict__ query,
    float* __restrict__ Y) {
  const int gid = blockIdx.x * 8 + (threadIdx.x >> 5);
  const int row0 = (gid / 16) * 32;
  const int col0 = (gid % 16) * 64;

  v8f acc[2][4];
  wave_gemm_32x64<true>(Cx, DMODEL, wo, DMODEL, row0, col0, DMODEL, acc);

  const int lane = threadIdx.x & 31, hlf = lane >> 4, mn = lane & 15;
#pragma unroll
  for (int i = 0; i < 2; ++i)
#pragma unroll
    for (int j = 0; j < 4; ++j)
#pragma unroll
      for (int r = 0; r < 8; ++r) {
        const int m = row0 + i * 16 + hlf * 8 + r;
        const int n = col0 + j * 16 + mn;
        Y[(size_t)m * DMODEL + n] = acc[i][j][r] + bo[n] + query[(size_t)m * DMODEL + n];
      }
}

// ---- Kernel 6: layernorm over last dim (1024) ------------------------------
__global__ void __launch_bounds__(256) ln_kernel(
    const float* __restrict__ Y, const float* __restrict__ gamma,
    const float* __restrict__ beta, float* __restrict__ out) {
  __shared__ float red[256];
  const float* p = Y + (size_t)blockIdx.x * DMODEL;
  const int t = threadIdx.x;
  float x[4], s = 0.f;
#pragma unroll
  for (int i = 0; i < 4; ++i) { x[i] = p[t + i * 256]; s += x[i]; }
  red[t] = s; __syncthreads();
  for (int o = 128; o > 0; o >>= 1) { if (t < o) red[t] += red[t + o]; __syncthreads(); }
  const float mu = red[0] * (1.0f / DMODEL);
  __syncthreads();
  float vs = 0.f;
#pragma unroll
  for (int i = 0; i < 4; ++i) { const float d = x[i] - mu; vs += d * d; }
  red[t] = vs; __syncthreads();
  for (int o = 128; o > 0; o >>= 1) { if (t < o) red[t] += red[t + o]; __syncthreads(); }
  const float rstd = rsqrtf(red[0] * (1.0f / DMODEL) + 1e-5f);
#pragma unroll
  for (int i = 0; i < 4; ++i) {
    const int n = t + i * 256;
    out[(size_t)blockIdx.x * DMODEL + n] = (x[i] - mu) * rstd * gamma[n] + beta[n];
  }
}

extern "C" void kernel_launch(void* const* d_in, const int* in_sizes, int n_in,
                              void* d_out, int out_size, void* d_ws, size_t ws_size,
                              hipStream_t stream) {
  const float* query = (const float*)d_in[0];
  const float* key_  = (const float*)d_in[1];
  const float* value = (const float*)d_in[2];
  const float* wq = (const float*)d_in[3];  const float* bq = (const float*)d_in[4];
  const float* wk = (const float*)d_in[5];  const float* bk = (const float*)d_in[6];
  const float* wv = (const float*)d_in[7];  const float* bv = (const float*)d_in[8];
  const float* wo = (const float*)d_in[9];  const float* bo = (const float*)d_in[10];
  const float* tb = (const float*)d_in[11];
  const float* at = (const float*)d_in[12];
  const float* g  = (const float*)d_in[13];
  const float* be = (const float*)d_in[14];

  const size_t TOK = (size_t)BATCH * SEQ * DMODEL;   // 4,194,304
  float* outln = (float*)d_out;
  float* attn  = outln + TOK;                        // 64M f32 attention matrix

  float* Qb = (float*)d_ws;      // (B,H,S,dk)
  float* Kb = Qb + TOK;
  float* Vb = Kb + TOK;
  float* Cx = Vb + TOK;          // bug-permuted ctx, viewed as (B*S, D)
  float* Yb = Qb;                // reuse Q slot after scores kernel

  qkv_kernel<<<768, 256, 0, stream>>>(query, key_, value, wq, bq, wk, bk, wv, bv,
                                      Qb, Kb, Vb);
  scores_kernel<<<4096, 256, 0, stream>>>(Qb, Kb, tb, at, attn);
  softmax_kernel<<<BATCH * NHEADS * SEQ, 256, 0, stream>>>(attn);
  ctx_kernel<<<256, 256, 0, stream>>>(attn, Vb, Cx);
  oproj_kernel<<<256, 256, 0, stream>>>(Cx, wo, bo, query, Yb);
  ln_kernel<<<BATCH * SEQ, 256, 0, stream>>>(Yb, g, be, outln);
}